// BRASKModel_8418135900642
// MI455X (gfx1250) — compile-verified
//
#include <hip/hip_runtime.h>
#include <math.h>

typedef float v2f __attribute__((ext_vector_type(2)));
typedef float v8f __attribute__((ext_vector_type(8)));
typedef int   v4i __attribute__((ext_vector_type(4)));

#define GAS __attribute__((address_space(1)))
#define LAS __attribute__((address_space(3)))

// ---------------- constants ----------------
#define BB 8
#define LL 256
#define HH 768
#define RR 24
#define TE 100
#define BL (BB * LL)          // 2048

// d_out offsets (floats)
#define O_FSS   0
#define O_FSE   2048
#define O_BOS   4096
#define O_BOE   6144
#define O_FSW   8192
#define O_FMASK 1581056
#define O_BOW   1583104
#define O_BMASK 3155968
#define O_FA    3158016
#define O_BA    3207168

// workspace offsets (floats)
#define W_FPAD  0                       // [BL,H]
#define W_BPAD  (W_FPAD + BL * HH)      // [BL,H]
#define W_FXW   (W_BPAD + BL * HH)      // [BL,H]
#define W_BXW   (W_FXW + BL * HH)       // [BL,H]
#define W_FRW   (W_BXW + BL * HH)       // [R,H]
#define W_BRW   (W_FRW + RR * HH)       // [R,H]
#define W_BRELP (W_BRW + RR * HH)       // [R,H]
#define W_FGW   (W_BRELP + RR * HH)     // [B,H]
#define W_BGW   (W_FGW + BB * HH)       // [B,H]
#define W_FV    (W_BGW + BB * HH)       // [B,L,R]
#define W_BV    (W_FV + BB * LL * RR)   // [B,L,R]

__device__ __forceinline__ float sigmoidf(float x) {
    return 1.0f / (1.0f + __expf(-x));
}

// ---------------- CDNA5 async copy helpers ----------------
#if __has_builtin(__builtin_amdgcn_global_load_async_to_lds_b128)
#define HAVE_ASYNC_LDS 1
#else
#define HAVE_ASYNC_LDS 0
#endif

__device__ __forceinline__ void async_cp16(const float* g, float* l) {
#if HAVE_ASYNC_LDS
    __builtin_amdgcn_global_load_async_to_lds_b128(
        (GAS v4i*)g, (LAS v4i*)l, 0, 0);
#else
    *(float4*)l = *(const float4*)g;
#endif
}

__device__ __forceinline__ void async_wait_all() {
#if HAVE_ASYNC_LDS && __has_builtin(__builtin_amdgcn_s_wait_asynccnt)
    __builtin_amdgcn_s_wait_asynccnt(0);
#endif
}

// ---------------- K1: start/end probabilities ----------------
__global__ __launch_bounds__(256)
void probs_kernel(const float* __restrict__ embs,
                  const float* __restrict__ w0, const float* __restrict__ b0,
                  const float* __restrict__ w1, const float* __restrict__ b1,
                  const float* __restrict__ w2, const float* __restrict__ b2,
                  const float* __restrict__ w3, const float* __restrict__ b3,
                  float* __restrict__ out) {
    int wid  = blockIdx.x * (blockDim.x >> 5) + (threadIdx.x >> 5); // (b*L + l)
    int lane = threadIdx.x & 31;
    const float* x = embs + (size_t)wid * HH;
    float s0 = 0.f, s1 = 0.f, s2 = 0.f, s3 = 0.f;
    for (int h = lane; h < HH; h += 32) {
        float e = x[h];
        s0 = fmaf(e, w0[h], s0);
        s1 = fmaf(e, w1[h], s1);
        s2 = fmaf(e, w2[h], s2);
        s3 = fmaf(e, w3[h], s3);
    }
    for (int off = 16; off; off >>= 1) {
        s0 += __shfl_xor(s0, off, 32);
        s1 += __shfl_xor(s1, off, 32);
        s2 += __shfl_xor(s2, off, 32);
        s3 += __shfl_xor(s3, off, 32);
    }
    if (lane == 0) {
        out[O_FSS + wid] = sigmoidf(s0 + b0[0]);
        out[O_FSE + wid] = sigmoidf(s1 + b1[0]);
        out[O_BOS + wid] = sigmoidf(s2 + b2[0]);
        out[O_BOE + wid] = sigmoidf(s3 + b3[0]);
    }
}

// ---------------- K2: greedy span pairing + compaction ----------------
__global__ __launch_bounds__(256)
void extract_kernel(const float* __restrict__ embs,
                    const float* __restrict__ probs,  // == d_out base
                    float* __restrict__ pad_f, float* __restrict__ pad_b,
                    float* __restrict__ d_out) {
    int b = blockIdx.x;
    int dir = blockIdx.y;   // 0 = forward(subject), 1 = backward(object)
    const float* sp = probs + (dir ? O_BOS : O_FSS) + b * LL;
    const float* ep = probs + (dir ? O_BOE : O_FSE) + b * LL;
    float* pad      = (dir ? pad_b : pad_f) + (size_t)b * LL * HH;
    float* maskOut  = d_out + (dir ? O_BMASK : O_FMASK) + b * LL;

    __shared__ unsigned char sm[LL], em[LL];
    __shared__ short srcS[LL], srcE[LL];
    __shared__ int cnt;

    int t = threadIdx.x;
    sm[t] = sp[t] > 0.5f;
    em[t] = ep[t] > 0.5f;
    __syncthreads();

    if (t == 0) {
        int last = -1, c = 0;
        for (int i = 0; i < LL; ++i) {
            if (sm[i]) {
                int j0 = (i > last + 1) ? i : (last + 1);
                int e = -1;
                for (int j = j0; j < LL; ++j) { if (em[j]) { e = j; break; } }
                if (e >= 0) { srcS[c] = (short)i; srcE[c] = (short)e; ++c; last = e; }
            }
        }
        cnt = c;
    }
    __syncthreads();
    int c = cnt;
    maskOut[t] = (t < c) ? 1.0f : 0.0f;

    int wave = t >> 5, lane = t & 31;
    for (int j = wave; j < LL; j += 8) {
        float* dst = pad + (size_t)j * HH;
        if (j < c) {
            const float* ps = embs + ((size_t)b * LL + srcS[j]) * HH;
            const float* pe = embs + ((size_t)b * LL + srcE[j]) * HH;
            for (int h = lane; h < HH; h += 32) dst[h] = 0.5f * (ps[h] + pe[h]);
        } else {
            for (int h = lane; h < HH; h += 32) dst[h] = 0.0f;
        }
    }
}

// ---------------- WMMA helpers ----------------
__device__ __forceinline__ v8f wmma_f32x4(v2f a, v2f b, v8f c) {
    return __builtin_amdgcn_wmma_f32_16x16x4_f32(false, a, false, b,
                                                 (short)0, c, false, false);
}

// ---------------- K3a: direct WMMA fp32 GEMM (any K%4==0) ----------------
__global__ __launch_bounds__(256)
void gemm_wmma(const float* __restrict__ A, const float* __restrict__ W,
               const float* __restrict__ bias, const float* __restrict__ rowScale,
               float* __restrict__ C, int M, int N, int K) {
    int wave = threadIdx.x >> 5;
    int lane = threadIdx.x & 31;
    int half = lane >> 4;
    int lm   = lane & 15;
    int n0 = (blockIdx.x * 8 + wave) * 16;
    int m0 = blockIdx.y * 16;
    if (n0 >= N) return;  // uniform within the wave

    int rowA = m0 + lm;
    if (rowA >= M) rowA = M - 1;   // clamp: only affects unstored C rows
    const float* Arow = A + (size_t)rowA * K;

    v8f acc = {};
    for (int k0 = 0; k0 < K; k0 += 4) {
        int ka = k0 + half * 2;
        v2f a;
        a.x = Arow[ka];
        a.y = Arow[ka + 1];
        v2f bfrag;
        bfrag.x = W[(size_t)ka * N + n0 + lm];
        bfrag.y = W[(size_t)(ka + 1) * N + n0 + lm];
        acc = wmma_f32x4(a, bfrag, acc);
    }

    float bn = bias ? bias[n0 + lm] : 0.0f;
#pragma unroll
    for (int v = 0; v < 8; ++v) {
        int row = m0 + v + 8 * half;
        if (row < M) {
            float o = acc[v] + bn;
            if (rowScale) o *= rowScale[row];
            C[(size_t)row * N + n0 + lm] = o;
        }
    }
}

// ---------------- K3b: LDS-staged, async double-buffered WMMA GEMM ----------
// Requires K % 32 == 0, N % 128 == 0. Block = 8 waves = one 16(M) x 128(N) tile.
// K chunks of 32 staged via GLOBAL_LOAD_ASYNC_TO_LDS_B128 (double buffered).
__global__ __launch_bounds__(256)
void gemm_wmma_lds(const float* __restrict__ A, const float* __restrict__ W,
                   const float* __restrict__ bias, const float* __restrict__ rowScale,
                   float* __restrict__ C, int M, int N, int K) {
    __shared__ float As[2][16 * 32];    // 4 KB
    __shared__ float Bs[2][32 * 128];   // 32 KB

    int t    = threadIdx.x;
    int wave = t >> 5;
    int lane = t & 31;
    int half = lane >> 4;
    int lm   = lane & 15;
    int n0 = blockIdx.x * 128;
    int m0 = blockIdx.y * 16;

    // B-chunk loader coords: 32 rows x 128 cols, 4 x b128 per thread
    int bRow = t >> 5;          // 0..7 (+8 per iter)
    int bCol = (t & 31) * 4;    // 0..124
    // A-chunk loader coords: 16 rows x 32 cols, 1 x b128 for threads 0..127
    int aRow = t >> 3;          // 0..31 (only 0..15 used)
    int aCol = (t & 7) * 4;     // 0..28
    int rowA = m0 + aRow;
    if (rowA >= M) rowA = M - 1;   // clamp: affects unstored C rows only
    const float* Abase = A + (size_t)rowA * K + aCol;
    const float* Wbase = W + n0 + bCol;

    int nChunks = K >> 5;

    // prologue: chunk 0 -> buffer 0
    {
        const float* wp = Wbase + (size_t)bRow * N;
#pragma unroll
        for (int i = 0; i < 4; ++i)
            async_cp16(wp + (size_t)(i * 8) * N, &Bs[0][(bRow + i * 8) * 128 + bCol]);
        if (t < 128) async_cp16(Abase, &As[0][aRow * 32 + aCol]);
    }

    v8f acc = {};
    for (int c = 0; c < nChunks; ++c) {
        int buf = c & 1;
        async_wait_all();          // my async fills of chunk c are complete
        __syncthreads();           // everyone's fills visible

        if (c + 1 < nChunks) {     // issue next chunk into the other buffer
            int k0 = (c + 1) << 5;
            const float* wp = Wbase + (size_t)(k0 + bRow) * N;
#pragma unroll
            for (int i = 0; i < 4; ++i)
                async_cp16(wp + (size_t)(i * 8) * N,
                           &Bs[buf ^ 1][(bRow + i * 8) * 128 + bCol]);
            if (t < 128) async_cp16(Abase + k0, &As[buf ^ 1][aRow * 32 + aCol]);
        }

        const float* as = &As[buf][lm * 32];
        const float* bs = &Bs[buf][wave * 16 + lm];
#pragma unroll
        for (int kk = 0; kk < 32; kk += 4) {
            int ka = kk + half * 2;
            v2f a;
            a.x = as[ka];
            a.y = as[ka + 1];
            v2f bfrag;
            bfrag.x = bs[ka * 128];
            bfrag.y = bs[(ka + 1) * 128];
            acc = wmma_f32x4(a, bfrag, acc);
        }
        __syncthreads();           // all reads of buf done before it is refilled
    }

    int nc = n0 + wave * 16 + lm;
    float bn = bias ? bias[nc] : 0.0f;
#pragma unroll
    for (int v = 0; v < 8; ++v) {
        int row = m0 + v + 8 * half;
        if (row < M) {
            float o = acc[v] + bn;
            if (rowScale) o *= rowScale[row];
            C[(size_t)row * N + nc] = o;
        }
    }
}

// ---------------- K4: additive-attention energies ----------------
__global__ __launch_bounds__(256)
void energy_kernel(const float* __restrict__ xW, const float* __restrict__ gW,
                   const float* __restrict__ rW, const float* __restrict__ Vw,
                   const float* __restrict__ Vb1, float* __restrict__ vout) {
    int wid  = blockIdx.x * (blockDim.x >> 5) + (threadIdx.x >> 5); // (b*L + l)
    int lane = threadIdx.x & 31;
    int b = wid >> 8;

    const float* x = xW + (size_t)wid * HH;
    const float* g = gW + (size_t)b * HH;

    float xg[24], vv[24];
#pragma unroll
    for (int i = 0; i < 24; ++i) {
        int h = lane + i * 32;
        xg[i] = x[h] + g[h];
        vv[i] = Vw[h];
    }
    float Vb = Vb1[0];

    for (int r = 0; r < RR; ++r) {
        const float* rw = rW + (size_t)r * HH;
        float acc = 0.0f;
#pragma unroll
        for (int i = 0; i < 24; ++i)
            acc = fmaf(vv[i], tanhf(rw[lane + i * 32] + xg[i]), acc);
        for (int off = 16; off; off >>= 1) acc += __shfl_xor(acc, off, 32);
        if (lane == 0) vout[(size_t)wid * RR + r] = acc + Vb;
    }
}

// ---------------- K5: softmax over L per (b,r) ----------------
__global__ __launch_bounds__(256)
void softmax_kernel(const float* __restrict__ vin, float* __restrict__ aout) {
    int r = blockIdx.x, b = blockIdx.y;
    int l = threadIdx.x;
    __shared__ float red[LL];

    float val = vin[((size_t)b * LL + l) * RR + r];
    red[l] = val;
    __syncthreads();
    for (int s = 128; s; s >>= 1) {
        if (l < s) red[l] = fmaxf(red[l], red[l + s]);
        __syncthreads();
    }
    float m = red[0];
    __syncthreads();
    float e = __expf(val - m);
    red[l] = e;
    __syncthreads();
    for (int s = 128; s; s >>= 1) {
        if (l < s) red[l] += red[l + s];
        __syncthreads();
    }
    float sum = red[0];
    aout[((size_t)b * LL + l) * RR + r] = e / sum;
}

// ---------------- launch ----------------
extern "C" void kernel_launch(void* const* d_in, const int* in_sizes, int n_in,
                              void* d_out, int out_size, void* d_ws, size_t ws_size,
                              hipStream_t stream) {
    const float* h_gs = (const float*)d_in[0];   // [8,768]
    const float* embs = (const float*)d_in[1];   // [8,256,768]
    const float* fss_w = (const float*)d_in[2];  const float* fss_b = (const float*)d_in[3];
    const float* fse_w = (const float*)d_in[4];  const float* fse_b = (const float*)d_in[5];
    const float* bos_w = (const float*)d_in[6];  const float* bos_b = (const float*)d_in[7];
    const float* boe_w = (const float*)d_in[8];  const float* boe_b = (const float*)d_in[9];
    const float* V_w   = (const float*)d_in[10]; const float* V_b   = (const float*)d_in[11];
    const float* fWs_w = (const float*)d_in[12]; const float* fWs_b = (const float*)d_in[13];
    const float* bWs_w = (const float*)d_in[14]; const float* bWs_b = (const float*)d_in[15];
    const float* fWr_w = (const float*)d_in[16]; const float* fWr_b = (const float*)d_in[17];
    const float* fWg_w = (const float*)d_in[18]; const float* fWg_b = (const float*)d_in[19];
    const float* fWx_w = (const float*)d_in[20]; const float* fWx_b = (const float*)d_in[21];
    const float* bWr_w = (const float*)d_in[22]; const float* bWr_b = (const float*)d_in[23];
    const float* bWg_w = (const float*)d_in[24]; const float* bWg_b = (const float*)d_in[25];
    const float* bWx_w = (const float*)d_in[26]; const float* bWx_b = (const float*)d_in[27];
    const float* rpj_w = (const float*)d_in[28]; const float* rpj_b = (const float*)d_in[29];
    const float* rel_embs   = (const float*)d_in[30];  // [24,768]
    const float* rel_transe = (const float*)d_in[31];  // [24,100]

    float* out = (float*)d_out;
    float* ws  = (float*)d_ws;

    float* fpad  = ws + W_FPAD;
    float* bpad  = ws + W_BPAD;
    float* fxW   = ws + W_FXW;
    float* bxW   = ws + W_BXW;
    float* frW   = ws + W_FRW;
    float* brW   = ws + W_BRW;
    float* brelp = ws + W_BRELP;
    float* fgW   = ws + W_FGW;
    float* bgW   = ws + W_BGW;
    float* fv    = ws + W_FV;
    float* bv    = ws + W_BV;

    // 1) probabilities (4 heads) -> d_out
    probs_kernel<<<BL / 8, 256, 0, stream>>>(embs,
        fss_w, fss_b, fse_w, fse_b, bos_w, bos_b, boe_w, boe_b, out);

    // 2) greedy span extraction -> pads (ws) + masks (d_out)
    extract_kernel<<<dim3(BB, 2), 256, 0, stream>>>(embs, out, fpad, bpad, out);

    // 3) GEMMs: K=768 via async-LDS WMMA kernel, K=100 via direct WMMA kernel
    dim3 gBig(HH / 128, BL / 16);    // 2048x768
    dim3 gRel(HH / 128, 2);          // 24x768
    dim3 gG(HH / 128, 1);            // 8x768
    gemm_wmma_lds<<<gBig, 256, 0, stream>>>(embs, fWx_w, fWx_b, nullptr, fxW, BL, HH, HH);
    gemm_wmma_lds<<<gBig, 256, 0, stream>>>(embs, bWx_w, bWx_b, nullptr, bxW, BL, HH, HH);
    gemm_wmma_lds<<<gBig, 256, 0, stream>>>(fpad, fWs_w, fWs_b, out + O_FMASK, out + O_FSW, BL, HH, HH);
    gemm_wmma_lds<<<gBig, 256, 0, stream>>>(bpad, bWs_w, bWs_b, out + O_BMASK, out + O_BOW, BL, HH, HH);
    gemm_wmma<<<gRel, 256, 0, stream>>>(rel_transe, rpj_w, rpj_b, nullptr, brelp, RR, HH, TE);
    gemm_wmma_lds<<<gRel, 256, 0, stream>>>(rel_embs, fWr_w, fWr_b, nullptr, frW, RR, HH, HH);
    gemm_wmma_lds<<<gRel, 256, 0, stream>>>(brelp, bWr_w, bWr_b, nullptr, brW, RR, HH, HH);
    gemm_wmma_lds<<<gG, 256, 0, stream>>>(h_gs, fWg_w, fWg_b, nullptr, fgW, BB, HH, HH);
    gemm_wmma_lds<<<gG, 256, 0, stream>>>(h_gs, bWg_w, bWg_b, nullptr, bgW, BB, HH, HH);

    // 4) attention energies
    energy_kernel<<<BL / 8, 256, 0, stream>>>(fxW, fgW, frW, V_w, V_b, fv);
    energy_kernel<<<BL / 8, 256, 0, stream>>>(bxW, bgW, brW, V_w, V_b, bv);

    // 5) softmax over sequence dim -> f_A / b_A
    softmax_kernel<<<dim3(RR, BB), 256, 0, stream>>>(fv, out + O_FA);
    softmax_kernel<<<dim3(RR, BB), 256, 0, stream>>>(bv, out + O_BA);
}